// HyperGraphConvolution_60060822667745
// MI455X (gfx1250) — compile-verified
//
#include <hip/hip_runtime.h>

typedef float v2f __attribute__((ext_vector_type(2)));
typedef float v4f __attribute__((ext_vector_type(4)));
typedef float v8f __attribute__((ext_vector_type(8)));

#define A_DIM 128
#define B_DIM 128

// ---------------------------------------------------------------------------
// Native fp32 global atomic add (non-returning -> STOREcnt). Device scope is
// required: multiple workgroups scatter into the same output rows. S_ENDPGM
// does an implicit wait-idle, so untracked counters drain before wave exit.
// ---------------------------------------------------------------------------
__device__ __forceinline__ void atomic_add_f32_dev(float* addr, float val) {
    asm volatile("global_atomic_add_f32 %0, %1, off scope:SCOPE_DEV"
                 :
                 : "v"(addr), "v"(val)
                 : "memory");
}

// ---------------------------------------------------------------------------
// Kernel 1: out[n][j] = bias[j]   (vectorized float4 broadcast)
// ---------------------------------------------------------------------------
__global__ __launch_bounds__(256) void bias_init_kernel(float* __restrict__ out,
                                                        const float* __restrict__ bias,
                                                        int total_vec4) {
    int idx = blockIdx.x * blockDim.x + threadIdx.x;      // one float4 per thread
    if (idx >= total_vec4) return;
    int j4 = idx & (B_DIM / 4 - 1);                       // column group 0..31
    v4f b = *(const v4f*)(bias + j4 * 4);
    *(v4f*)(out + (size_t)idx * 4) = b;
}

// ---------------------------------------------------------------------------
// Kernel 2: HW = H @ W via V_WMMA_F32_16X16X4_F32 (full fp32 precision).
// One wave32 computes one 16x16 tile of HW; K-loop = 128/4 = 32 WMMAs.
//
// fp32 A 16x4 fragment: lane L -> row M = L%16, K = 2*(L/16) + {0,1}  (v2f)
// fp32 B 4x16 fragment: lane L -> col N = L%16, K = 2*(L/16) + {0,1}  (v2f)
// fp32 C/D 16x16:       VGPR r -> M = r + 8*(L/16), N = L%16          (v8f)
// ---------------------------------------------------------------------------
__global__ __launch_bounds__(256) void gemm_hw_wmma_kernel(const float* __restrict__ H,
                                                           const float* __restrict__ W,
                                                           float* __restrict__ HW,
                                                           int n_nodes,
                                                           int n_tiles_m) {
    const int wave_in_blk = threadIdx.x >> 5;
    const int lane        = threadIdx.x & 31;
    const int wave_gid    = blockIdx.x * 8 + wave_in_blk;

    const int tile_m = wave_gid >> 3;            // 8 column tiles (128/16)
    const int tile_n = wave_gid & 7;
    if (tile_m >= n_tiles_m) return;

    const int m0   = tile_m * 16;
    const int n0   = tile_n * 16;
    const int half = lane >> 4;                  // 0 or 1 (selects K pair)
    const int l16  = lane & 15;
    const int koff = half * 2;

    // Clamp row for tail safety (EXEC must stay all-ones for WMMA; garbage
    // rows are simply never stored).
    int rowA = m0 + l16;
    if (rowA >= n_nodes) rowA = n_nodes - 1;
    const int colB = n0 + l16;

    const float* aptr = H + (size_t)rowA * A_DIM + koff;   // 8B aligned (koff even)
    const float* bptr = W + (size_t)koff * B_DIM + colB;

    v8f acc = {};
#pragma unroll
    for (int k = 0; k < A_DIM; k += 4) {
        v2f a = *(const v2f*)(aptr + k);                   // H[rowA][k+koff .. +1]
        v2f b;
        b.x = bptr[(size_t)k * B_DIM];                     // W[k+koff  ][colB]
        b.y = bptr[(size_t)(k + 1) * B_DIM];               // W[k+koff+1][colB]
        // 8 args: (neg_a, A, neg_b, B, c_mod, C, reuse_a, reuse_b)
        acc = __builtin_amdgcn_wmma_f32_16x16x4_f32(
                  false, a, false, b, (short)0, acc, false, false);
    }

    // Store D: VGPR r -> row m0 + r + 8*half, col colB
    const int rbase = m0 + half * 8;
#pragma unroll
    for (int r = 0; r < 8; ++r) {
        int row = rbase + r;
        if (row < n_nodes) HW[(size_t)row * B_DIM + colB] = acc[r];
    }
}

// ---------------------------------------------------------------------------
// Kernel 3: per-edge gather/scale/scatter.
// One wave32 per edge; each lane owns 4 consecutive output columns:
//   out[r][j] += v * HW[c][j]   via native global_atomic_add_f32.
// HW and out are both L2-resident (25.6 MB each vs 192 MB L2), so this is
// L2-atomic-throughput bound, not HBM bound.
// Edge metadata index is routed through readfirstlane so the compiler can
// prove wave-uniformity and use scalar (SMEM) loads for rows/cols/vals.
// ---------------------------------------------------------------------------
__global__ __launch_bounds__(256) void edge_scatter_kernel(const int* __restrict__ rows,
                                                           const int* __restrict__ cols,
                                                           const float* __restrict__ vals,
                                                           const float* __restrict__ HW,
                                                           float* __restrict__ out,
                                                           int n_edges) {
    const int wave = (int)((blockIdx.x * (size_t)blockDim.x + threadIdx.x) >> 5);
    const int lane = threadIdx.x & 31;
    if (wave >= n_edges) return;

    const int e = __builtin_amdgcn_readfirstlane(wave);    // wave-uniform edge id

    const int   r = rows[e];
    const int   c = cols[e];
    const float v = vals[e];

    const float* src = HW  + (size_t)c * B_DIM + lane * 4;
    float*       dst = out + (size_t)r * B_DIM + lane * 4;

    v4f m = *(const v4f*)src;                   // global_load_b128 (512B / wave)
    atomic_add_f32_dev(dst + 0, v * m.x);       // 4x global_atomic_add_f32
    atomic_add_f32_dev(dst + 1, v * m.y);
    atomic_add_f32_dev(dst + 2, v * m.z);
    atomic_add_f32_dev(dst + 3, v * m.w);
}

// ---------------------------------------------------------------------------
extern "C" void kernel_launch(void* const* d_in, const int* in_sizes, int n_in,
                              void* d_out, int out_size, void* d_ws, size_t ws_size,
                              hipStream_t stream) {
    const int*   edge_rows = (const int*)  d_in[0];
    const int*   edge_cols = (const int*)  d_in[1];
    const float* edge_vals = (const float*)d_in[2];
    const float* H         = (const float*)d_in[3];
    const float* W         = (const float*)d_in[4];
    const float* bias      = (const float*)d_in[5];

    const int n_edges = in_sizes[0];
    const int n_nodes = in_sizes[3] / A_DIM;

    float* out = (float*)d_out;
    float* HW  = (float*)d_ws;        // n_nodes*128 floats = 25.6 MB scratch

    // 1) out = broadcast(bias)
    const int total4 = n_nodes * (B_DIM / 4);
    bias_init_kernel<<<(total4 + 255) / 256, 256, 0, stream>>>(out, bias, total4);

    // 2) HW = H @ W  (fp32 WMMA)
    const int n_tiles_m = (n_nodes + 15) / 16;
    const int n_waves   = n_tiles_m * (B_DIM / 16);
    gemm_hw_wmma_kernel<<<(n_waves + 7) / 8, 256, 0, stream>>>(H, W, HW, n_nodes, n_tiles_m);

    // 3) out[r] += v * HW[c]  (atomic scatter, one wave per edge)
    const int s_blocks = (n_edges + 7) / 8;   // 8 waves (edges) per 256-thread block
    edge_scatter_kernel<<<s_blocks, 256, 0, stream>>>(edge_rows, edge_cols, edge_vals,
                                                      HW, out, n_edges);
}